// PointCloudTransformerAE_fullmodel_5987184411284
// MI455X (gfx1250) — compile-verified
//
#include <hip/hip_runtime.h>
#include <hip/hip_bf16.h>

typedef __attribute__((ext_vector_type(2))) float v2f;
typedef __attribute__((ext_vector_type(8))) float v8f;

#define FEAT 64
#define BSEG 128
#define EPSLN 1e-5f

// ---------- monotone float <-> uint encoding (for atomicMax on floats) ----------
__device__ __forceinline__ unsigned int enc_f32(float f) {
    int i = __float_as_int(f);
    return (i < 0) ? ~((unsigned int)i) : (((unsigned int)i) | 0x80000000u);
}
__device__ __forceinline__ float dec_f32(unsigned int u) {
    unsigned int i = (u & 0x80000000u) ? (u & 0x7fffffffu) : ~u;
    return __uint_as_float(i);
}
// enc(-inf) = ~0xFF800000 = 0x007FFFFF
#define ENC_NEG_INF 0x007FFFFFu

// ---------- init pooled accumulator ----------
__global__ void initk(unsigned int* __restrict__ p, int n, unsigned int val) {
    int i = blockIdx.x * blockDim.x + threadIdx.x;
    if (i < n) p[i] = val;
}

// ---------- streaming segment-max (batch ids are sorted) ----------
// Block = 256 threads. Thread t: fq = t&15 (which float4 of the 64-float row),
// r = t>>4 (point sub-index). Each block owns a contiguous chunk of 4096 points:
// 256 iterations x 16 points. b128 loads, fully coalesced (4KB/iter/block).
__global__ void segmax_kernel(const float4* __restrict__ feat4,
                              const int*    __restrict__ batch,
                              unsigned int* __restrict__ pooled,
                              int npts) {
    const int t  = threadIdx.x;
    const int fq = t & 15;       // float4 index within a 64-float feature row
    const int r  = t >> 4;       // 0..15
    const long long base = (long long)blockIdx.x * 4096;

    int curSeg = -1;
    float4 cm = make_float4(0.f, 0.f, 0.f, 0.f);

    for (int it = 0; it < 256; ++it) {
        long long p = base + (long long)it * 16 + r;
        if (p >= npts) break;
        // prefetch ~8 iterations (128 points) ahead
        if (p + 128 < npts)
            __builtin_prefetch(&feat4[(p + 128) * 16 + fq], 0, 1);

        int seg = batch[p];
        float4 v = feat4[p * 16 + fq];
        if (seg != curSeg) {
            if (curSeg >= 0) {
                unsigned int* dst = pooled + (size_t)curSeg * FEAT + fq * 4;
                atomicMax(dst + 0, enc_f32(cm.x));
                atomicMax(dst + 1, enc_f32(cm.y));
                atomicMax(dst + 2, enc_f32(cm.z));
                atomicMax(dst + 3, enc_f32(cm.w));
            }
            curSeg = seg;
            cm = v;
        } else {
            cm.x = fmaxf(cm.x, v.x);
            cm.y = fmaxf(cm.y, v.y);
            cm.z = fmaxf(cm.z, v.z);
            cm.w = fmaxf(cm.w, v.w);
        }
    }
    if (curSeg >= 0) {
        unsigned int* dst = pooled + (size_t)curSeg * FEAT + fq * 4;
        atomicMax(dst + 0, enc_f32(cm.x));
        atomicMax(dst + 1, enc_f32(cm.y));
        atomicMax(dst + 2, enc_f32(cm.z));
        atomicMax(dst + 3, enc_f32(cm.w));
    }
}

// ---------- LayerNorm(64): one wave32 per row, 2 values/lane ----------
__global__ void layernorm_kernel(const unsigned int* __restrict__ pooled,
                                 const float* __restrict__ g,
                                 const float* __restrict__ b,
                                 float* __restrict__ xout) {
    const int row = blockIdx.x;
    const int t   = threadIdx.x;               // 0..31
    const int i0  = row * FEAT + 2 * t;

    float v0 = dec_f32(pooled[i0]);
    float v1 = dec_f32(pooled[i0 + 1]);

    float s  = v0 + v1;
    float ss = v0 * v0 + v1 * v1;
    #pragma unroll
    for (int off = 16; off > 0; off >>= 1) {
        s  += __shfl_xor(s,  off, 32);
        ss += __shfl_xor(ss, off, 32);
    }
    float mu  = s * (1.0f / FEAT);
    float var = ss * (1.0f / FEAT) - mu * mu;
    float inv = rsqrtf(var + EPSLN);

    xout[i0]     = (v0 - mu) * inv * g[2 * t]     + b[2 * t];
    xout[i0 + 1] = (v1 - mu) * inv * g[2 * t + 1] + b[2 * t + 1];
}

// ---------- fp32 WMMA GEMM: C[M,N] = relu?(A[M,K] @ W[K,N] + bias) ----------
// One wave per 16x16 tile. grid = (N/16, M/16), block = 32.
// ISA f32 layouts:
//  A 16x4 : lane&15 = M row; half-wave picks K pair; VGPR0=K+kl, VGPR1=K+kl+1
//  B 4x16 : lane&15 = N col; half-wave picks K pair; VGPR0=K+kl, VGPR1=K+kl+1
//  C 16x16: lane&15 = N col; VGPR i -> M = i + 8*(lane>>4)
__global__ void gemm_wmma_kernel(const float* __restrict__ A,
                                 const float* __restrict__ W,
                                 const float* __restrict__ bias,
                                 float* __restrict__ C,
                                 int K, int N, int relu) {
    const int lane = threadIdx.x & 31;
    const int l16  = lane & 15;
    const int hlf  = lane >> 4;       // 0 or 1
    const int kl   = hlf * 2;         // K sub-offset within a step of 4
    const int mt   = blockIdx.y;
    const int nt   = blockIdx.x;
    const int m    = mt * 16 + l16;   // A row handled by this lane
    const int n    = nt * 16 + l16;   // B/C column handled by this lane

    const float* __restrict__ arow = A + (size_t)m * K;

    v8f acc = {};
    for (int k = 0; k < K; k += 4) {
        v2f a, bv;
        a.x  = arow[k + kl];
        a.y  = arow[k + kl + 1];
        bv.x = W[(size_t)(k + kl)     * N + n];
        bv.y = W[(size_t)(k + kl + 1) * N + n];
        acc = __builtin_amdgcn_wmma_f32_16x16x4_f32(
                  /*neg_a=*/false, a, /*neg_b=*/false, bv,
                  /*c_mod=*/(short)0, acc, /*reuse_a=*/false, /*reuse_b=*/false);
    }

    const float bn = bias[n];
    #pragma unroll
    for (int i = 0; i < 8; ++i) {
        int row = mt * 16 + i + 8 * hlf;
        float v = acc[i] + bn;
        if (relu) v = fmaxf(v, 0.0f);
        C[(size_t)row * N + n] = v;
    }
}

extern "C" void kernel_launch(void* const* d_in, const int* in_sizes, int n_in,
                              void* d_out, int out_size, void* d_ws, size_t ws_size,
                              hipStream_t stream) {
    const float* feat   = (const float*)d_in[0];   // [npts, 64]
    const int*   batch  = (const int*)  d_in[1];   // [npts] sorted ids
    const float* ln_g   = (const float*)d_in[2];   // [64]
    const float* ln_b   = (const float*)d_in[3];   // [64]
    const float* proj_w = (const float*)d_in[4];   // [64, 512]
    const float* proj_b = (const float*)d_in[5];   // [512]
    const float* w1     = (const float*)d_in[6];   // [512, 256]
    const float* b1     = (const float*)d_in[7];   // [256]
    const float* w2     = (const float*)d_in[8];   // [256, 512]
    const float* b2     = (const float*)d_in[9];   // [512]
    const float* w3     = (const float*)d_in[10];  // [512, 6144]
    const float* b3     = (const float*)d_in[11];  // [6144]
    float* out = (float*)d_out;                    // [128 * 6144]

    const int npts = in_sizes[1];

    // workspace layout (all 4-byte elems)
    unsigned int* pooled = (unsigned int*)d_ws;                 // 128*64
    float* xln    = (float*)d_ws + BSEG * FEAT;                 // 128*64
    float* latent = xln    + BSEG * FEAT;                       // 128*512
    float* h1     = latent + BSEG * 512;                        // 128*256
    float* h2     = h1     + BSEG * 256;                        // 128*512

    // 1) init pooled to enc(-inf)
    {
        int n = BSEG * FEAT;
        initk<<<(n + 255) / 256, 256, 0, stream>>>(pooled, n, ENC_NEG_INF);
    }
    // 2) streaming segment-max (the 256 MB bandwidth-bound stage)
    {
        int blocks = (npts + 4095) / 4096;
        segmax_kernel<<<blocks, 256, 0, stream>>>((const float4*)feat, batch, pooled, npts);
    }
    // 3) LayerNorm -> xln
    layernorm_kernel<<<BSEG, 32, 0, stream>>>(pooled, ln_g, ln_b, xln);
    // 4) latent = xln @ proj_w + proj_b            [128,64]x[64,512]
    gemm_wmma_kernel<<<dim3(512 / 16, BSEG / 16), 32, 0, stream>>>(xln, proj_w, proj_b, latent, 64, 512, 0);
    // 5) h1 = relu(latent @ w1 + b1)               [128,512]x[512,256]
    gemm_wmma_kernel<<<dim3(256 / 16, BSEG / 16), 32, 0, stream>>>(latent, w1, b1, h1, 512, 256, 1);
    // 6) h2 = relu(h1 @ w2 + b2)                   [128,256]x[256,512]
    gemm_wmma_kernel<<<dim3(512 / 16, BSEG / 16), 32, 0, stream>>>(h1, w2, b2, h2, 256, 512, 1);
    // 7) out = h2 @ w3 + b3                        [128,512]x[512,6144]
    gemm_wmma_kernel<<<dim3(6144 / 16, BSEG / 16), 32, 0, stream>>>(h2, w3, b3, out, 512, 6144, 0);
}